// GAT_38079180046923
// MI455X (gfx1250) — compile-verified
//
#include <hip/hip_runtime.h>
#include <math.h>

#define N_NODES 20000
#define N_EDGES 640000
#define HID     64
#define HEADS   3
#define HH      192                      // HEADS * HID
#define TOTAL   (N_EDGES + N_NODES)      // rows of ef / srcf / dstf = 660000

typedef float v2f __attribute__((ext_vector_type(2)));
typedef float v8f __attribute__((ext_vector_type(8)));
typedef int   eidx_t;                    // edge_index element type (harness: integer -> int*)

// ---------------------------------------------------------------------------
// fe = 10*linspace(0,1,64)^2 ;  v[0..63]=fe@commW_top, v[64..]=fe@commW_bot,
// v[128..]=fe@tollW, v[192..]=fe@edgeW   (rank-1 collapse of input embeddings)
// ---------------------------------------------------------------------------
__global__ void gat_prevec(const float* __restrict__ commW,
                           const float* __restrict__ tollW,
                           const float* __restrict__ edgeW,
                           float* __restrict__ v) {
  int c = threadIdx.x;                   // 0..63
  float s1 = 0.f, s2 = 0.f, s3 = 0.f, s4 = 0.f;
  for (int j = 0; j < HID; ++j) {
    float t  = (float)j * (1.0f / 63.0f);
    float fe = 10.0f * t * t;
    s1 += fe * commW[j * HID + c];
    s2 += fe * commW[(HID + j) * HID + c];
    s3 += fe * tollW[j * HID + c];
    s4 += fe * edgeW[j * HID + c];
  }
  v[c] = s1; v[HID + c] = s2; v[2 * HID + c] = s3; v[3 * HID + c] = s4;
}

__global__ void gat_node_h(const float* __restrict__ x, const float* __restrict__ v,
                           const float* __restrict__ comm_b, const float* __restrict__ toll_b,
                           float* __restrict__ h0) {
  int t = blockIdx.x * blockDim.x + threadIdx.x;
  if (t >= N_NODES * HID) return;
  int n = t / HID, c = t - n * HID;
  float mask = x[n * 5 + 0], x1 = x[n * 5 + 1], x2 = x[n * 5 + 2], x3 = x[n * 5 + 3];
  float comm = fmaxf(x1 * v[c] + x2 * v[HID + c] + comm_b[c], 0.f);
  float toll = fmaxf(x3 * v[2 * HID + c] + toll_b[c], 0.f);
  h0[t] = comm * (1.f - mask) + toll * mask;
}

// eemb into ef[0..E); atomic accumulate loop_attr into ef[E..E+N); degree count
__global__ void gat_edge(const float* __restrict__ edge_attr, const float* __restrict__ v,
                         const float* __restrict__ edge_b, const eidx_t* __restrict__ dst,
                         float* __restrict__ ef, float* __restrict__ deg) {
  int t = blockIdx.x * blockDim.x + threadIdx.x;
  if (t >= N_EDGES * HID) return;
  int e = t / HID, c = t - e * HID;
  float val = fmaxf(edge_attr[e] * v[3 * HID + c] + edge_b[c], 0.f);
  ef[(size_t)e * HID + c] = val;
  int d = dst[e];
  atomicAdd(&ef[((size_t)N_EDGES + d) * HID + c], val);
  if (c == 0) atomicAdd(&deg[d], 1.f);
}

__global__ void gat_loopnorm(float* __restrict__ ef, const float* __restrict__ deg) {
  int t = blockIdx.x * blockDim.x + threadIdx.x;
  if (t >= N_NODES * HID) return;
  int n = t / HID;
  ef[((size_t)N_EDGES + n) * HID + (t - n * HID)] *= 1.f / fmaxf(deg[n], 1.f);
}

// ---------------------------------------------------------------------------
// fp32 WMMA GEMM, K a compile-time constant (full unroll), 2 row-tiles/wave:
//   C(32r x 16c) = A(rows x K) @ W(K x M) + bias
// grid = (rows/32, M/16), block = 32 (one wave).  Two independent
// accumulators break the serial WMMA D->C chain and share B fragments.
// ---------------------------------------------------------------------------
template <int K>
__global__ __launch_bounds__(32)
void gat_gemm16(const float* __restrict__ A, const float* __restrict__ W,
                const float* __restrict__ bias, float* __restrict__ C, int M) {
  const int lane = threadIdx.x;
  const int ln = lane & 15, half = lane >> 4;
  const int row0 = blockIdx.x * 32;
  const int col  = blockIdx.y * 16 + ln;
  float bv = bias[col];
  v8f acc0, acc1;
#pragma unroll
  for (int r = 0; r < 8; ++r) { acc0[r] = bv; acc1[r] = bv; }
  const float* arow0 = A + (size_t)(row0 + ln) * K;
  const float* arow1 = A + (size_t)(row0 + 16 + ln) * K;
#pragma unroll
  for (int k0 = 0; k0 < K; k0 += 4) {
    int k = k0 + half * 2;
    v2f a0; a0.x = arow0[k];               a0.y = arow0[k + 1];
    v2f a1; a1.x = arow1[k];               a1.y = arow1[k + 1];
    v2f b;  b.x  = W[(size_t)k * M + col]; b.y  = W[(size_t)(k + 1) * M + col];
    acc0 = __builtin_amdgcn_wmma_f32_16x16x4_f32(false, a0, false, b, (short)0, acc0,
                                                 false, false);
    acc1 = __builtin_amdgcn_wmma_f32_16x16x4_f32(false, a1, false, b, (short)0, acc1,
                                                 false, false);
  }
#pragma unroll
  for (int r = 0; r < 8; ++r) {
    C[(size_t)(row0 + r + half * 8) * M + col]      = acc0[r];
    C[(size_t)(row0 + 16 + r + half * 8) * M + col] = acc1[r];
  }
}

// ---------------------------------------------------------------------------
// Fused score kernel: per 16-row block of (E+N):
//   m_tile = gather(xl[src]) + gather(xr[dst])  (accumulator init)
//          + ef_rows(16x64) @ We(64x192)        (12 ctiles x 16 fp32 WMMAs)
//   s[row,h] = sum_c lrelu(m)*att[h,c]  -> ex = exp(s) ; denom[dst,h] += ex
// ---------------------------------------------------------------------------
__global__ __launch_bounds__(32)
void gat_score(const float* __restrict__ ef, const float* __restrict__ We,
               const float* __restrict__ att, const float* __restrict__ xl,
               const float* __restrict__ xr, const eidx_t* __restrict__ src,
               const eidx_t* __restrict__ dst, float* __restrict__ exv,
               float* __restrict__ denom) {
  __shared__ int   sidx[16], didx[16];
  __shared__ float sblk[48];               // [row][head]
  const int lane = threadIdx.x;
  const int ln = lane & 15, half = lane >> 4;
  const int row0 = blockIdx.x * 16;
  if (lane < 16) {
    int r = row0 + lane, s_, d_;
    if (r < N_EDGES) { s_ = src[r]; d_ = dst[r]; } else { s_ = d_ = r - N_EDGES; }
    sidx[lane] = s_; didx[lane] = d_;
    sblk[lane] = 0.f; sblk[lane + 16] = 0.f; sblk[lane + 32] = 0.f;
  }
  __syncthreads();

  v8f acc[12];
#pragma unroll
  for (int ct = 0; ct < 12; ++ct) {
    int col = ct * 16 + ln;
#pragma unroll
    for (int r = 0; r < 8; ++r) {
      int m = r + half * 8;
      acc[ct][r] = xl[(size_t)sidx[m] * HH + col] + xr[(size_t)didx[m] * HH + col];
    }
  }

  const float* arow = ef + (size_t)(row0 + ln) * HID;
#pragma unroll
  for (int kk = 0; kk < HID; kk += 4) {
    int k0 = kk + half * 2;
    v2f a; a.x = arow[k0]; a.y = arow[k0 + 1];
#pragma unroll
    for (int ct = 0; ct < 12; ++ct) {
      v2f b;
      b.x = We[(size_t)k0 * HH + ct * 16 + ln];
      b.y = We[(size_t)(k0 + 1) * HH + ct * 16 + ln];
      acc[ct] = __builtin_amdgcn_wmma_f32_16x16x4_f32(false, a, false, b, (short)0,
                                                      acc[ct], false, false);
    }
  }

#pragma unroll
  for (int ct = 0; ct < 12; ++ct) {
    int head = ct >> 2;
    float attv = att[head * HID + (ct & 3) * 16 + ln];
#pragma unroll
    for (int r = 0; r < 8; ++r) {
      float vv = acc[ct][r];
      vv = (vv > 0.f) ? vv : 0.2f * vv;        // leaky_relu(0.2)
      float p = vv * attv;
      p += __shfl_xor(p, 1);
      p += __shfl_xor(p, 2);
      p += __shfl_xor(p, 4);
      p += __shfl_xor(p, 8);                   // sum over 16 cols within half-wave
      if (ln == 0) atomicAdd(&sblk[(r + half * 8) * 3 + head], p);
    }
  }
  __syncthreads();
  if (lane < 16) {
    int r = row0 + lane, d_ = didx[lane];
#pragma unroll
    for (int h = 0; h < 3; ++h) {
      float e_ = expf(sblk[lane * 3 + h]);     // smax skipped: alpha invariant
      exv[(size_t)r * 3 + h] = e_;
      atomicAdd(&denom[(size_t)d_ * 3 + h], e_);
    }
  }
}

__global__ void gat_aggregate(const float* __restrict__ exv, const float* __restrict__ denom,
                              const float* __restrict__ xl, const eidx_t* __restrict__ src,
                              const eidx_t* __restrict__ dst, float* __restrict__ g) {
  int t = blockIdx.x * blockDim.x + threadIdx.x;
  if (t >= TOTAL * HH) return;               // 126,720,000 < 2^31
  int r = t / HH, c = t - r * HH;
  int s_, d_;
  if (r < N_EDGES) { s_ = src[r]; d_ = dst[r]; } else { s_ = d_ = r - N_EDGES; }
  int h = c >> 6;
  float alpha = exv[(size_t)r * 3 + h] / (denom[(size_t)d_ * 3 + h] + 1e-16f);
  atomicAdd(&g[(size_t)d_ * HH + c], alpha * xl[(size_t)s_ * HH + c]);
}

__global__ void gat_node_update(const float* __restrict__ g, const float* __restrict__ bias,
                                const float* __restrict__ cb, const float* __restrict__ tb,
                                const float* __restrict__ x, float* __restrict__ hn) {
  int t = blockIdx.x * blockDim.x + threadIdx.x;
  if (t >= N_NODES * HH) return;
  int n = t / HH, c = t - n * HH;
  float mask = x[n * 5];
  float val = g[t] + bias[c] + cb[t] * (1.f - mask) + tb[t] * mask;
  hn[t] = fmaxf(val, 0.f);
}

// --------------------------- layer 3 (Hh=1, C=1) ---------------------------
__global__ void gat_lin3(const float* __restrict__ h2, const float* __restrict__ Wl3,
                         const float* __restrict__ bl3, const float* __restrict__ Wr3,
                         const float* __restrict__ br3, float* __restrict__ xl3,
                         float* __restrict__ xr3) {
  int n = blockIdx.x * blockDim.x + threadIdx.x;
  if (n >= N_NODES) return;
  float sl = bl3[0], sr = br3[0];
  const float* hr = h2 + (size_t)n * HH;
  for (int k = 0; k < HH; ++k) { float hv = hr[k]; sl += hv * Wl3[k]; sr += hv * Wr3[k]; }
  xl3[n] = sl; xr3[n] = sr;
}

__global__ void gat_score3(const float* __restrict__ ef, const float* __restrict__ We3,
                           const float* __restrict__ att3, const float* __restrict__ xl3,
                           const float* __restrict__ xr3, const eidx_t* __restrict__ src,
                           const eidx_t* __restrict__ dst, float* __restrict__ ex3,
                           float* __restrict__ denom3) {
  int r = blockIdx.x * blockDim.x + threadIdx.x;
  if (r >= TOTAL) return;
  int s_, d_;
  if (r < N_EDGES) { s_ = src[r]; d_ = dst[r]; } else { s_ = d_ = r - N_EDGES; }
  float ew = 0.f;
  const float* er = ef + (size_t)r * HID;
  for (int k = 0; k < HID; ++k) ew += er[k] * We3[k];
  float m = xl3[s_] + xr3[d_] + ew;
  m = (m > 0.f) ? m : 0.2f * m;
  float e_ = expf(m * att3[0]);
  ex3[r] = e_;
  atomicAdd(&denom3[d_], e_);
}

__global__ void gat_agg3(const float* __restrict__ ex3, const float* __restrict__ denom3,
                         const float* __restrict__ xl3, const eidx_t* __restrict__ src,
                         const eidx_t* __restrict__ dst, float* __restrict__ g3) {
  int r = blockIdx.x * blockDim.x + threadIdx.x;
  if (r >= TOTAL) return;
  int s_, d_;
  if (r < N_EDGES) { s_ = src[r]; d_ = dst[r]; } else { s_ = d_ = r - N_EDGES; }
  float alpha = ex3[r] / (denom3[d_] + 1e-16f);
  atomicAdd(&g3[d_], alpha * xl3[s_]);
}

__global__ void gat_final(const float* __restrict__ g3, const float* __restrict__ bias3,
                          const float* __restrict__ c3buf, const float* __restrict__ t3buf,
                          const float* __restrict__ x, float* __restrict__ out) {
  int t = blockIdx.x * blockDim.x + threadIdx.x;
  if (t >= N_NODES * HID) return;
  int n = t / HID;
  float mask = x[n * 5];
  float val = g3[n] + bias3[0] + c3buf[t] * (1.f - mask) + t3buf[t] * mask;
  out[t] = val * mask;
}

// ---------------------------------------------------------------------------
extern "C" void kernel_launch(void* const* d_in, const int* in_sizes, int n_in,
                              void* d_out, int out_size, void* d_ws, size_t ws_size,
                              hipStream_t stream) {
  (void)in_sizes; (void)n_in; (void)out_size; (void)ws_size;
  const float*  x         = (const float*)d_in[0];
  const eidx_t* eidx      = (const eidx_t*)d_in[1];
  const eidx_t* src       = eidx;
  const eidx_t* dst       = eidx + N_EDGES;
  const float*  edge_attr = (const float*)d_in[2];
  const float *comm_W=(const float*)d_in[3], *comm_b=(const float*)d_in[4];
  const float *toll_W=(const float*)d_in[5], *toll_b=(const float*)d_in[6];
  const float *edge_W=(const float*)d_in[7], *edge_b=(const float*)d_in[8];
  const float *Wl1=(const float*)d_in[9],  *bl1=(const float*)d_in[10];
  const float *Wr1=(const float*)d_in[11], *br1=(const float*)d_in[12];
  const float *We1=(const float*)d_in[13], *att1=(const float*)d_in[14], *bias1=(const float*)d_in[15];
  const float *Wl2=(const float*)d_in[16], *bl2=(const float*)d_in[17];
  const float *Wr2=(const float*)d_in[18], *br2=(const float*)d_in[19];
  const float *We2=(const float*)d_in[20], *att2=(const float*)d_in[21], *bias2=(const float*)d_in[22];
  const float *Wl3=(const float*)d_in[23], *bl3=(const float*)d_in[24];
  const float *Wr3=(const float*)d_in[25], *br3=(const float*)d_in[26];
  const float *We3=(const float*)d_in[27], *att3=(const float*)d_in[28], *bias3=(const float*)d_in[29];
  const float *c1W=(const float*)d_in[30], *c1b=(const float*)d_in[31];
  const float *c2W=(const float*)d_in[32], *c2b=(const float*)d_in[33];
  const float *c3W=(const float*)d_in[34], *c3b=(const float*)d_in[35];
  const float *t1W=(const float*)d_in[36], *t1b=(const float*)d_in[37];
  const float *t2W=(const float*)d_in[38], *t2b=(const float*)d_in[39];
  const float *t3W=(const float*)d_in[40], *t3b=(const float*)d_in[41];

  float* ws = (float*)d_ws;
  size_t off = 0;
  auto alloc = [&](size_t nf) { float* p = ws + off; off += nf; return p; };
  float* v      = alloc(4 * HID);
  float* h0     = alloc((size_t)N_NODES * HID);
  float* ef     = alloc((size_t)TOTAL * HID);
  float* xl     = alloc((size_t)N_NODES * HH);
  float* xr     = alloc((size_t)N_NODES * HH);
  float* cbuf   = alloc((size_t)N_NODES * HH);
  float* tbuf   = alloc((size_t)N_NODES * HH);
  float* g      = alloc((size_t)N_NODES * HH);
  float* h1     = alloc((size_t)N_NODES * HH);
  float* h2     = alloc((size_t)N_NODES * HH);
  float* exv    = alloc((size_t)TOTAL * 3);
  float* denom  = alloc((size_t)N_NODES * 3);
  float* deg    = alloc(N_NODES);
  float* xl3    = alloc(N_NODES);
  float* xr3    = alloc(N_NODES);
  float* ex3    = alloc(TOTAL);
  float* denom3 = alloc(N_NODES);
  float* g3     = alloc(N_NODES);
  float* c3buf  = alloc((size_t)N_NODES * HID);
  float* t3buf  = alloc((size_t)N_NODES * HID);

  // ---- input embeddings + self-loop attributes ----
  hipMemsetAsync(deg, 0, N_NODES * sizeof(float), stream);
  hipMemsetAsync(ef + (size_t)N_EDGES * HID, 0, (size_t)N_NODES * HID * sizeof(float), stream);
  gat_prevec<<<1, HID, 0, stream>>>(comm_W, toll_W, edge_W, v);
  gat_node_h<<<(N_NODES * HID + 255) / 256, 256, 0, stream>>>(x, v, comm_b, toll_b, h0);
  gat_edge<<<(N_EDGES * HID + 255) / 256, 256, 0, stream>>>(edge_attr, v, edge_b, dst, ef, deg);
  gat_loopnorm<<<(N_NODES * HID + 255) / 256, 256, 0, stream>>>(ef, deg);

  dim3 gA(N_NODES / 32, HH / 16);            // (625, 12)
  dim3 gC(N_NODES / 32, HID / 16);           // (625, 4)
  const int aggBlocks = (TOTAL * HH + 255) / 256;

  // ---- GAT layer 1 (F_in = 64) ----
  gat_gemm16<HID><<<gA, 32, 0, stream>>>(h0, Wl1, bl1, xl,   HH);
  gat_gemm16<HID><<<gA, 32, 0, stream>>>(h0, Wr1, br1, xr,   HH);
  gat_gemm16<HID><<<gA, 32, 0, stream>>>(h0, c1W, c1b, cbuf, HH);
  gat_gemm16<HID><<<gA, 32, 0, stream>>>(h0, t1W, t1b, tbuf, HH);
  hipMemsetAsync(denom, 0, N_NODES * 3 * sizeof(float), stream);
  gat_score<<<TOTAL / 16, 32, 0, stream>>>(ef, We1, att1, xl, xr, src, dst, exv, denom);
  hipMemsetAsync(g, 0, (size_t)N_NODES * HH * sizeof(float), stream);
  gat_aggregate<<<aggBlocks, 256, 0, stream>>>(exv, denom, xl, src, dst, g);
  gat_node_update<<<(N_NODES * HH + 255) / 256, 256, 0, stream>>>(g, bias1, cbuf, tbuf, x, h1);

  // ---- GAT layer 2 (F_in = 192) ----
  gat_gemm16<HH><<<gA, 32, 0, stream>>>(h1, Wl2, bl2, xl,   HH);
  gat_gemm16<HH><<<gA, 32, 0, stream>>>(h1, Wr2, br2, xr,   HH);
  gat_gemm16<HH><<<gA, 32, 0, stream>>>(h1, c2W, c2b, cbuf, HH);
  gat_gemm16<HH><<<gA, 32, 0, stream>>>(h1, t2W, t2b, tbuf, HH);
  hipMemsetAsync(denom, 0, N_NODES * 3 * sizeof(float), stream);
  gat_score<<<TOTAL / 16, 32, 0, stream>>>(ef, We2, att2, xl, xr, src, dst, exv, denom);
  hipMemsetAsync(g, 0, (size_t)N_NODES * HH * sizeof(float), stream);
  gat_aggregate<<<aggBlocks, 256, 0, stream>>>(exv, denom, xl, src, dst, g);
  gat_node_update<<<(N_NODES * HH + 255) / 256, 256, 0, stream>>>(g, bias2, cbuf, tbuf, x, h2);

  // ---- GAT layer 3 (single head/channel) + skip + output ----
  gat_gemm16<HH><<<gC, 32, 0, stream>>>(h2, c3W, c3b, c3buf, HID);
  gat_gemm16<HH><<<gC, 32, 0, stream>>>(h2, t3W, t3b, t3buf, HID);
  gat_lin3<<<(N_NODES + 255) / 256, 256, 0, stream>>>(h2, Wl3, bl3, Wr3, br3, xl3, xr3);
  hipMemsetAsync(denom3, 0, N_NODES * sizeof(float), stream);
  hipMemsetAsync(g3, 0, N_NODES * sizeof(float), stream);
  gat_score3<<<(TOTAL + 255) / 256, 256, 0, stream>>>(ef, We3, att3, xl3, xr3, src, dst, ex3, denom3);
  gat_agg3<<<(TOTAL + 255) / 256, 256, 0, stream>>>(ex3, denom3, xl3, src, dst, g3);
  gat_final<<<(N_NODES * HID + 255) / 256, 256, 0, stream>>>(g3, bias3, c3buf, t3buf, x,
                                                             (float*)d_out);
}